// MultiHeadAttention_19782619365484
// MI455X (gfx1250) — compile-verified
//
#include <hip/hip_runtime.h>
#include <cstdint>

#define Bb 2
#define Ll 2048
#define Dd 1024
#define Hh 16
#define DKk 64
#define DVv 64
#define LN_EPS 1e-6f
#define KC 32   // K-chunk staged in LDS

typedef __attribute__((ext_vector_type(2))) float v2f;
typedef __attribute__((ext_vector_type(8))) float v8f;

static __device__ __forceinline__ v8f wmma_f32(v2f a, v2f b, v8f c) {
  // D = A(16x4) * B(4x16) + C, fp32, V_WMMA_F32_16X16X4_F32
  return __builtin_amdgcn_wmma_f32_16x16x4_f32(
      /*neg_a=*/false, a, /*neg_b=*/false, b,
      /*c_mod=*/(short)0, c, /*reuse_a=*/false, /*reuse_b=*/false);
}

// gfx1250 async global->LDS copy (GV mode), tracked by ASYNCcnt.
// vdst = VGPR with LDS byte offset, vaddr = 64-bit global address.
static __device__ __forceinline__ void async_copy_b128(const float* gp, const float* lp) {
  unsigned lds_off = (unsigned)(uintptr_t)lp;  // low 32 bits of generic addr == LDS offset
  asm volatile("global_load_async_to_lds_b128 %0, %1, off"
               :: "v"(lds_off), "v"(gp)
               : "memory");
}

// ---------------------------------------------------------------------------
// Staged fp32 WMMA GEMM:  Y = scale * (A[MxK] @ B[KxN]) (+ residual)
// lda == K, ldb == N. 128-thread block = 4 waves; block tile 64(M) x 64(N).
// B K-chunks (KC x 64) are double-buffered in LDS via async-to-LDS b128 copies
// shared by all 4 waves; A fragments are pre-loaded to registers per chunk.
// mode 0: plain row-major (+ optional residual)
// mode 1: head-split scatter  (m=b*L+l, n=h*64+d) -> Y[((b*H+h)*L+l)*64+d]
// mode 2: ctx scatter (batched by blockIdx.z = b*H+h) -> Y[(b*L+m)*D + h*64+n]
// ---------------------------------------------------------------------------
__global__ __launch_bounds__(128)
void wmma_gemm_staged(const float* __restrict__ A, const float* __restrict__ Bm,
                      const float* __restrict__ residual, float* __restrict__ Y,
                      int M, int N, int K,
                      unsigned long long aStride, unsigned long long bStride,
                      float scale, int mode) {
  __shared__ float bufB[2][KC * 64];   // 2 x 8 KB

  const int tid  = threadIdx.x;
  const int lane = tid & 31;
  const int w    = tid >> 5;
  const int half = lane >> 4;          // K pair {0,1} vs {2,3}
  const int r16  = lane & 15;
  const int bz   = blockIdx.z;
  const int n0   = blockIdx.y * 64;
  const int mBase = blockIdx.x * 64 + w * 16;

  const float* Ab   = A + (size_t)bz * aStride;
  const float* Bb2  = Bm + (size_t)bz * bStride + n0;
  const float* arow = Ab + (size_t)(mBase + r16) * K + 2 * half;

  v8f c[4] = {};
  const int nChunks = K / KC;

  // Prefetch chunk 0 into buffer 0 (4 async b128 per thread).
  for (int i = tid * 4; i < KC * 64; i += 128 * 4) {
    int kr = i >> 6, col = i & 63;
    async_copy_b128(Bb2 + (size_t)kr * N + col, &bufB[0][i]);
  }

  for (int j = 0; j < nChunks; ++j) {
    const int buf = j & 1;
    if (j + 1 < nChunks) {
      // Prefetch next chunk into the other buffer, then wait for current one.
      const float* src = Bb2 + (size_t)(j + 1) * KC * N;
      for (int i = tid * 4; i < KC * 64; i += 128 * 4) {
        int kr = i >> 6, col = i & 63;
        async_copy_b128(src + (size_t)kr * N + col, &bufB[buf ^ 1][i]);
      }
      asm volatile("s_wait_asynccnt 0x4" ::: "memory");
    } else {
      asm volatile("s_wait_asynccnt 0x0" ::: "memory");
    }
    __syncthreads();

    // Pre-load this chunk's A fragments (8 x b64, clause-able).
    const float* ap = arow + j * KC;
    v2f areg[KC / 4];
#pragma unroll
    for (int s = 0; s < KC / 4; ++s)
      areg[s] = *(const v2f*)(ap + 4 * s);

    const float* lb = &bufB[buf][0];
#pragma unroll
    for (int s = 0; s < KC / 4; ++s) {
      const int kk = 4 * s + 2 * half;
#pragma unroll
      for (int t = 0; t < 4; ++t) {
        v2f b;
        b.x = lb[kk * 64 + t * 16 + r16];
        b.y = lb[(kk + 1) * 64 + t * 16 + r16];
        c[t] = wmma_f32(areg[s], b, c[t]);
      }
    }
    __syncthreads();
  }

#pragma unroll
  for (int t = 0; t < 4; ++t) {
#pragma unroll
    for (int r = 0; r < 8; ++r) {
      int m = mBase + r + 8 * half;
      int n = n0 + t * 16 + r16;
      float val = c[t][r] * scale;
      if (mode == 0) {
        if (residual) val += residual[(size_t)m * N + n];
        Y[(size_t)m * N + n] = val;
      } else if (mode == 1) {
        int b = m / Ll, l = m - b * Ll;
        int h = n >> 6, dd = n & 63;
        Y[(((size_t)b * Hh + h) * Ll + l) * 64 + dd] = val;
      } else {
        int b = bz / Hh, h = bz - (bz / Hh) * Hh;
        Y[((size_t)b * Ll + m) * Dd + h * 64 + n] = val;
      }
    }
  }
}

// ---------------------------------------------------------------------------
// Attention scores: S[bh] = qh[bh] @ kh[bh]^T (qh pre-scaled by 1/sqrt(DK)).
// B = K^T so the (k,k+1) fragment pair is contiguous in kh rows (b64 loads).
// A fragments (whole K=64) pre-loaded to registers; raw scores -> attn region.
// ---------------------------------------------------------------------------
__global__ __launch_bounds__(32)
void score_kernel(const float* __restrict__ qh,
                  const float* __restrict__ kh,
                  float* __restrict__ attn) {
  const int lane = threadIdx.x & 31;
  const int m0 = blockIdx.x * 16;   // query tile
  const int n0 = blockIdx.y * 64;   // key strip
  const int bh = blockIdx.z;
  const int half = lane >> 4;
  const int r16  = lane & 15;

  const float* qrow  = qh + ((size_t)bh * Ll + (m0 + r16)) * DKk + 2 * half;
  const float* kbase = kh + (size_t)bh * Ll * DKk;

  v2f aq[DKk / 4];
#pragma unroll
  for (int s = 0; s < DKk / 4; ++s)
    aq[s] = *(const v2f*)(qrow + 4 * s);

  v8f c[4] = {};
#pragma unroll
  for (int s = 0; s < DKk / 4; ++s) {
    const int k = 4 * s;
#pragma unroll
    for (int t = 0; t < 4; ++t) {
      const float* kr = kbase + (size_t)(n0 + t * 16 + r16) * DKk + k + 2 * half;
      v2f b = *(const v2f*)kr;   // contiguous -> b64
      c[t] = wmma_f32(aq[s], b, c[t]);
    }
  }

#pragma unroll
  for (int t = 0; t < 4; ++t)
#pragma unroll
    for (int r = 0; r < 8; ++r) {
      int m = m0 + r + 8 * half;
      int n = n0 + t * 16 + r16;
      attn[((size_t)bh * Ll + m) * Ll + n] = c[t][r];
    }
}

// ---------------------------------------------------------------------------
// Masked row softmax in place over attn rows of length L. Block = 256 = 8 waves.
// ---------------------------------------------------------------------------
__global__ __launch_bounds__(256)
void softmax_mask_kernel(float* __restrict__ attn,
                         const int* __restrict__ mask) {
  const int row = blockIdx.x;        // (b*H + h)*L + qi
  const int bh = row / Ll;
  const int qi = row - bh * Ll;
  const int b  = bh / Hh;

  float* p = attn + (size_t)row * Ll;
  const int* mrow = mask + ((size_t)b * Ll + qi) * Ll;

  __shared__ float redA[8];
  __shared__ float redB[8];
  const int tid = threadIdx.x;

  float mx = -3.0e38f;
  for (int j = tid; j < Ll; j += 256) {
    float s = (mrow[j] == 0) ? -1.0e9f : p[j];
    mx = fmaxf(mx, s);
  }
  for (int o = 16; o; o >>= 1) mx = fmaxf(mx, __shfl_xor(mx, o, 32));
  if ((tid & 31) == 0) redA[tid >> 5] = mx;
  __syncthreads();
  if (tid == 0) {
    float v = redA[0];
    for (int i = 1; i < 8; ++i) v = fmaxf(v, redA[i]);
    redA[0] = v;
  }
  __syncthreads();
  mx = redA[0];

  float sum = 0.0f;
  for (int j = tid; j < Ll; j += 256) {
    float s = (mrow[j] == 0) ? -1.0e9f : p[j];
    float e = expf(s - mx);
    p[j] = e;
    sum += e;
  }
  for (int o = 16; o; o >>= 1) sum += __shfl_xor(sum, o, 32);
  if ((tid & 31) == 0) redB[tid >> 5] = sum;
  __syncthreads();
  if (tid == 0) {
    float v = 0.0f;
    for (int i = 0; i < 8; ++i) v += redB[i];
    redB[0] = v;
  }
  __syncthreads();
  const float inv = 1.0f / redB[0];

  for (int j = tid; j < Ll; j += 256) p[j] *= inv;
}

// ---------------------------------------------------------------------------
// LayerNorm per row of D=1024 (biased variance, matching jnp.var).
// ---------------------------------------------------------------------------
__global__ __launch_bounds__(256)
void layernorm_kernel(const float* __restrict__ X,
                      const float* __restrict__ g,
                      const float* __restrict__ beta,
                      float* __restrict__ out) {
  const int row = blockIdx.x;
  const float* x = X + (size_t)row * Dd;
  float* o = out + (size_t)row * Dd;
  const int tid = threadIdx.x;

  __shared__ float rs[8];
  __shared__ float rq[8];
  float s = 0.0f, q = 0.0f;
  for (int j = tid; j < Dd; j += 256) {
    float v = x[j];
    s += v;
    q += v * v;
  }
  for (int o2 = 16; o2; o2 >>= 1) {
    s += __shfl_xor(s, o2, 32);
    q += __shfl_xor(q, o2, 32);
  }
  if ((tid & 31) == 0) { rs[tid >> 5] = s; rq[tid >> 5] = q; }
  __syncthreads();
  if (tid == 0) {
    float a = 0.0f, bq = 0.0f;
    for (int i = 0; i < 8; ++i) { a += rs[i]; bq += rq[i]; }
    rs[0] = a; rq[0] = bq;
  }
  __syncthreads();
  const float mean = rs[0] / (float)Dd;
  const float var  = rq[0] / (float)Dd - mean * mean;
  const float inv  = rsqrtf(var + LN_EPS);
  for (int j = tid; j < Dd; j += 256)
    o[j] = (x[j] - mean) * inv * g[j] + beta[j];
}

// ---------------------------------------------------------------------------
extern "C" void kernel_launch(void* const* d_in, const int* in_sizes, int n_in,
                              void* d_out, int out_size, void* d_ws, size_t ws_size,
                              hipStream_t stream) {
  (void)in_sizes; (void)n_in; (void)out_size; (void)ws_size;

  const float* q    = (const float*)d_in[0];
  const float* k    = (const float*)d_in[1];
  const float* v    = (const float*)d_in[2];
  const int*   mask = (const int*)d_in[3];
  const float* w_q  = (const float*)d_in[4];
  const float* w_k  = (const float*)d_in[5];
  const float* w_v  = (const float*)d_in[6];
  const float* w_fc = (const float*)d_in[7];
  const float* ln_g = (const float*)d_in[8];
  const float* ln_b = (const float*)d_in[9];

  float* out_main = (float*)d_out;                       // (B, L, D)
  float* attn     = out_main + (size_t)Bb * Ll * Dd;     // (B, H, L, L)

  // Workspace: qh | kh | vh (each B*H*L*64 = 4M floats).
  // ctx aliases kh, pre-LN y aliases qh (both dead after score_kernel).
  float* ws = (float*)d_ws;
  const size_t headElems = (size_t)Bb * Hh * Ll * 64;
  float* qh  = ws;
  float* kh  = qh + headElems;
  float* vh  = kh + headElems;
  float* ctx = kh;   // reuse
  float* y   = qh;   // reuse

  const dim3 blk128(128);

  // QKV projections (head-split layout); fold 1/sqrt(DK)=0.125 into qh.
  const dim3 gproj((Bb * Ll) / 64, (Hh * DKk) / 64, 1);
  wmma_gemm_staged<<<gproj, blk128, 0, stream>>>(q, w_q, nullptr, qh,
                                                 Bb * Ll, Hh * DKk, Dd, 0, 0, 0.125f, 1);
  wmma_gemm_staged<<<gproj, blk128, 0, stream>>>(k, w_k, nullptr, kh,
                                                 Bb * Ll, Hh * DKk, Dd, 0, 0, 1.0f, 1);
  wmma_gemm_staged<<<gproj, blk128, 0, stream>>>(v, w_v, nullptr, vh,
                                                 Bb * Ll, Hh * DVv, Dd, 0, 0, 1.0f, 1);

  // Raw scores -> attn region.
  score_kernel<<<dim3(Ll / 16, Ll / 64, Bb * Hh), dim3(32), 0, stream>>>(qh, kh, attn);

  // Masked softmax in place (this is also the returned attn).
  softmax_mask_kernel<<<dim3(Bb * Hh * Ll), dim3(256), 0, stream>>>(attn, mask);

  // attn @ V -> ctx (token-major), batched over bh via blockIdx.z.
  wmma_gemm_staged<<<dim3(Ll / 64, 1, Bb * Hh), blk128, 0, stream>>>(
      attn, vh, nullptr, ctx, Ll, DVv, Ll,
      (unsigned long long)Ll * Ll, (unsigned long long)Ll * DVv, 1.0f, 2);

  // Output projection + residual -> y.
  wmma_gemm_staged<<<dim3((Bb * Ll) / 64, Dd / 64, 1), blk128, 0, stream>>>(
      ctx, w_fc, q, y, Bb * Ll, Dd, Dd, 0, 0, 1.0f, 0);

  // LayerNorm -> final out.
  layernorm_kernel<<<dim3(Bb * Ll), dim3(256), 0, stream>>>(y, ln_g, ln_b, out_main);
}